// HeadModel_816043786336
// MI455X (gfx1250) — compile-verified
//
#include <hip/hip_runtime.h>
#include <hip/hip_bf16.h>
#include <cstdint>

typedef __attribute__((ext_vector_type(2))) float v2f;
typedef __attribute__((ext_vector_type(8))) float v8f;

#define HID     128
#define NNODES  100000
#define NEDGES  1600000
#define NGRAPHS 256
#define BN_EPS  1e-5f

// ---------------------------------------------------------------------------
// Fused GEMM: [K|Q|V|S] = x @ [Wk|Wq|Wv|Ws] + [bk|bq|bv|bias]
// grid.x = NNODES/16 (row tiles), block = 256 (8 waves).
// Wave w covers a 16x64 slab: matrix (w>>1), column base (w&1)*64.
// Uses V_WMMA_F32_16X16X4_F32, K-loop of 32 steps over d_in=128.
// ---------------------------------------------------------------------------
__global__ void gemm_kqvs_wmma(const float* __restrict__ X,
    const float* __restrict__ Wk, const float* __restrict__ bk,
    const float* __restrict__ Wq, const float* __restrict__ bq,
    const float* __restrict__ Wv, const float* __restrict__ bv,
    const float* __restrict__ Ws, const float* __restrict__ bs,
    float* __restrict__ K, float* __restrict__ Q,
    float* __restrict__ V, float* __restrict__ S)
{
    const int wave = threadIdx.x >> 5;
    const int lane = threadIdx.x & 31;
    const int m0   = blockIdx.x << 4;
    const int mat  = wave >> 1;
    const float* __restrict__ W = (mat == 0) ? Wk : (mat == 1) ? Wq : (mat == 2) ? Wv : Ws;
    const float* __restrict__ B = (mat == 0) ? bk : (mat == 1) ? bq : (mat == 2) ? bv : bs;
    float*       __restrict__ Y = (mat == 0) ? K  : (mat == 1) ? Q  : (mat == 2) ? V  : S;
    const int n0 = (wave & 1) << 6;

    const int l16 = lane >> 4;     // 0: K pair {0,1} / M row; 1: K pair {2,3}
    const int mlo = lane & 15;     // M (A frag) / N (B frag, C frag)
    const float* __restrict__ xrow = X + (size_t)(m0 + mlo) * HID + (l16 << 1);
    const int brow = l16 << 1;

    v8f acc[4] = {};
    for (int k0 = 0; k0 < HID; k0 += 4) {
        v2f a;
        a.x = xrow[k0];
        a.y = xrow[k0 + 1];
        const float* __restrict__ wp = W + (size_t)(k0 + brow) * HID + n0 + mlo;
#pragma unroll
        for (int t = 0; t < 4; ++t) {
            v2f b;
            b.x = wp[t * 16];
            b.y = wp[t * 16 + HID];
            acc[t] = __builtin_amdgcn_wmma_f32_16x16x4_f32(
                false, a, false, b, (short)0, acc[t], false, false);
        }
    }
#pragma unroll
    for (int t = 0; t < 4; ++t) {
        const int col = n0 + t * 16 + mlo;
        const float bb = B[col];
#pragma unroll
        for (int j = 0; j < 8; ++j) {
            const int r = m0 + j + (l16 << 3);
            Y[(size_t)r * HID + col] = acc[t][j] + bb;
        }
    }
}

// ---------------------------------------------------------------------------
// Edge phase: one wave32 per edge; lane handles 4 contiguous features.
// msg = sigmoid(k[dst] + q[src]) * v[src], scatter-added into OUT[dst]
// with hardware fp32 global atomics (no return -> STOREcnt path).
// ---------------------------------------------------------------------------
__global__ void edge_gate_scatter(const int* __restrict__ ei,
    const float* __restrict__ K, const float* __restrict__ Q,
    const float* __restrict__ V, float* __restrict__ OUT)
{
    const int e    = (blockIdx.x << 3) + (threadIdx.x >> 5);
    const int lane = threadIdx.x & 31;
    const int src  = ei[e];
    const int dst  = ei[NEDGES + e];

    const float4 kk = ((const float4*)(K + (size_t)dst * HID))[lane];
    const float4 qq = ((const float4*)(Q + (size_t)src * HID))[lane];
    const float4 vv = ((const float4*)(V + (size_t)src * HID))[lane];

    float4 m;
    m.x = vv.x / (1.0f + __expf(-(kk.x + qq.x)));
    m.y = vv.y / (1.0f + __expf(-(kk.y + qq.y)));
    m.z = vv.z / (1.0f + __expf(-(kk.z + qq.z)));
    m.w = vv.w / (1.0f + __expf(-(kk.w + qq.w)));

    float* o = OUT + (size_t)dst * HID + (lane << 2);
    unsafeAtomicAdd(o + 0, m.x);
    unsafeAtomicAdd(o + 1, m.y);
    unsafeAtomicAdd(o + 2, m.z);
    unsafeAtomicAdd(o + 3, m.w);
}

// ---------------------------------------------------------------------------
// Fused ReLU + BN statistics: thread t owns column (t&127), strided rows.
// Writes relu(h) in place, accumulates column sum / sumsq locally, LDS
// combine, then 2 atomics per column per block into stats[0..255].
// ---------------------------------------------------------------------------
__global__ void relu_stats(float* __restrict__ H, float* __restrict__ stats, int n)
{
    const int col = threadIdx.x & (HID - 1);
    const int sub = threadIdx.x >> 7;  // 0 or 1
    float s = 0.0f, s2 = 0.0f;
    for (int r = blockIdx.x * 2 + sub; r < n; r += gridDim.x * 2) {
        float v = H[(size_t)r * HID + col];
        v = fmaxf(v, 0.0f);
        H[(size_t)r * HID + col] = v;
        s  += v;
        s2 += v * v;
    }
    __shared__ float ls[256];
    __shared__ float ls2[256];
    ls[threadIdx.x]  = s;
    ls2[threadIdx.x] = s2;
    __syncthreads();
    if (sub == 0) {
        unsafeAtomicAdd(&stats[col],        ls[col]  + ls[col + HID]);
        unsafeAtomicAdd(&stats[HID + col],  ls2[col] + ls2[col + HID]);
    }
}

// BN normalize in place: y = (x - mean) * rsqrt(var+eps) * gamma + beta
__global__ void bn_apply(float* __restrict__ H, const float* __restrict__ stats,
                         const float* __restrict__ gamma, const float* __restrict__ beta,
                         int n)
{
    const size_t idx = (size_t)blockIdx.x * blockDim.x + threadIdx.x;
    const int col = (int)(idx & (HID - 1));
    const float invn = 1.0f / (float)n;
    const float mean = stats[col] * invn;
    const float var  = stats[HID + col] * invn - mean * mean;
    const float sc   = rsqrtf(var + BN_EPS) * gamma[col];
    H[idx] = (H[idx] - mean) * sc + beta[col];
}

// Graph pooling: one wave per node, float4 per lane, fp32 atomics into pooled.
__global__ void pool_scatter(const float* __restrict__ H, const int* __restrict__ batch,
                             float* __restrict__ pooled)
{
    const int node = (blockIdx.x << 3) + (threadIdx.x >> 5);
    if (node >= NNODES) return;
    const int lane = threadIdx.x & 31;
    const int g = batch[node];
    const float4 v = ((const float4*)(H + (size_t)node * HID))[lane];
    float* p = pooled + (size_t)g * HID + (lane << 2);
    unsafeAtomicAdd(p + 0, v.x);
    unsafeAtomicAdd(p + 1, v.y);
    unsafeAtomicAdd(p + 2, v.z);
    unsafeAtomicAdd(p + 3, v.w);
}

// Head: out[g] = pooled[g,:] . Wout + bout   (256 graphs, one thread each)
__global__ void head_out(const float* __restrict__ pooled,
                         const float* __restrict__ Wout, const float* __restrict__ bout,
                         float* __restrict__ out)
{
    const int g = threadIdx.x;
    if (g >= NGRAPHS) return;
    float acc = bout[0];
    for (int c = 0; c < HID; ++c)
        acc = fmaf(pooled[(size_t)g * HID + c], Wout[c], acc);
    out[g] = acc;
}

__global__ void zero_f32(float* __restrict__ p, int n)
{
    const int i = blockIdx.x * blockDim.x + threadIdx.x;
    if (i < n) p[i] = 0.0f;
}

// ---------------------------------------------------------------------------
// Input order (setup_inputs dict, leaves in insertion order):
//  0: x [100000,128] f32
//  1..8  : layer0  Wk,bk,Wq,bq,Wv,bv,Ws,bias
//  9..16 : layer1  Wk,bk,Wq,bq,Wv,bv,Ws,bias
// 17,18 : bn0 gamma,beta   19,20 : bn1 gamma,beta
// 21,22 : out W [128,1], b [1]
// 23: edge_index [2,1600000] int   24: batch [100000] int
// ---------------------------------------------------------------------------
extern "C" void kernel_launch(void* const* d_in, const int* in_sizes, int n_in,
                              void* d_out, int out_size, void* d_ws, size_t ws_size,
                              hipStream_t stream)
{
    (void)in_sizes; (void)n_in; (void)out_size; (void)ws_size;

    const float* x = (const float*)d_in[0];
    const float* L[2][8];
    for (int l = 0; l < 2; ++l)
        for (int j = 0; j < 8; ++j)
            L[l][j] = (const float*)d_in[1 + l * 8 + j];
    const float* gamma[2] = {(const float*)d_in[17], (const float*)d_in[19]};
    const float* beta[2]  = {(const float*)d_in[18], (const float*)d_in[20]};
    const float* Wout = (const float*)d_in[21];
    const float* bout = (const float*)d_in[22];
    const int*   ei    = (const int*)d_in[23];
    const int*   batch = (const int*)d_in[24];

    float* ws = (float*)d_ws;
    const size_t NF = (size_t)NNODES * HID;
    float* Kb     = ws;
    float* Qb     = ws + NF;
    float* Vb     = ws + 2 * NF;
    float* bufA   = ws + 3 * NF;
    float* bufB   = ws + 4 * NF;
    float* stats  = ws + 5 * NF;          // 256 floats: [sum|sumsq]
    float* pooled = ws + 5 * NF + 256;    // 256*128 floats

    const float* hin = x;
    float* houts[2] = {bufA, bufB};
    for (int l = 0; l < 2; ++l) {
        float* hout = houts[l];
        zero_f32<<<1, 256, 0, stream>>>(stats, 2 * HID);
        // hout = x@Ws + bias (residual branch lands directly in agg buffer)
        gemm_kqvs_wmma<<<NNODES / 16, 256, 0, stream>>>(hin,
            L[l][0], L[l][1], L[l][2], L[l][3], L[l][4], L[l][5], L[l][6], L[l][7],
            Kb, Qb, Vb, hout);
        edge_gate_scatter<<<NEDGES / 8, 256, 0, stream>>>(ei, Kb, Qb, Vb, hout);
        relu_stats<<<512, 256, 0, stream>>>(hout, stats, NNODES);
        bn_apply<<<(NNODES * HID) / 256, 256, 0, stream>>>(hout, stats, gamma[l], beta[l], NNODES);
        hin = hout;
    }
    zero_f32<<<(NGRAPHS * HID) / 256, 256, 0, stream>>>(pooled, NGRAPHS * HID);
    pool_scatter<<<NNODES / 8, 256, 0, stream>>>(bufB, batch, pooled);
    head_out<<<1, 256, 0, stream>>>(pooled, Wout, bout, (float*)d_out);
}